// Ins_17669495455827
// MI455X (gfx1250) — compile-verified
//
#include <hip/hip_runtime.h>
#include <hip/hip_bf16.h>
#include <math.h>

typedef __attribute__((ext_vector_type(2))) float v2f;
typedef __attribute__((ext_vector_type(8))) float v8f;

#define CAND_CAP 4096
#define SORT_N   4096

// ws layout (in u32 units):
//  [0,3072)      hist[pass][sel][256]   (3 sels * 4 passes * 256 bins)
//  [3072,3075)   prefix[3]   (accumulated threshold key)
//  [3075,3078)   kRem[3]
//  [3078,3081)   cnt[3]      (compaction counters)
//  [3082,27658)  cand[3][CAND_CAP][2]   (key, idx)
//  [27658,...)   selIdx[3][K]
#define WS_HIST   0
#define WS_PREFIX 3072
#define WS_KREM   3075
#define WS_CNT    3078
#define WS_CAND   3082
#define WS_SEL    27658

__device__ __forceinline__ unsigned monoKey(float f) {
  unsigned u = __float_as_uint(f);
  return (u & 0x80000000u) ? ~u : (u | 0x80000000u);
}

__global__ void initKernel(unsigned* ws, int K) {
  int t = threadIdx.x;
  for (int i = t; i < 3072; i += blockDim.x) ws[WS_HIST + i] = 0u;
  if (t < 3) {
    ws[WS_PREFIX + t] = 0u;
    ws[WS_KREM + t]   = (unsigned)K;
    ws[WS_CNT + t]    = 0u;
  }
}

__global__ void histKernel(const float* __restrict__ A, const int* __restrict__ blp,
                           unsigned* ws, int N, int pass) {
  __shared__ unsigned lh[768];
  int t = threadIdx.x;
  for (int i = t; i < 768; i += blockDim.x) lh[i] = 0u;
  __syncthreads();
  int bl = blp[0] & 1;
  int shift = 8 * (3 - pass);
  unsigned p0 = ws[WS_PREFIX + 0], p1 = ws[WS_PREFIX + 1], p2 = ws[WS_PREFIX + 2];
  for (int i = blockIdx.x * blockDim.x + t; i < N; i += gridDim.x * blockDim.x) {
    float ain  = A[2 * i + bl];
    float aout = A[2 * i + (1 - bl)];
    unsigned keys[3];
    keys[0] = monoKey(ain);
    keys[1] = monoKey(-ain);
    keys[2] = monoKey(aout);
    unsigned pref[3] = {p0, p1, p2};
#pragma unroll
    for (int s = 0; s < 3; ++s) {
      bool ok;
      if (pass == 0) ok = true;
      else {
        int sh8 = shift + 8;
        ok = ((keys[s] >> sh8) == (pref[s] >> sh8));
      }
      if (ok) atomicAdd(&lh[s * 256 + ((keys[s] >> shift) & 255u)], 1u);
    }
  }
  __syncthreads();
  unsigned* gh = ws + WS_HIST + (pass * 3) * 256;
  for (int i = t; i < 768; i += blockDim.x) {
    unsigned v = lh[i];
    if (v) atomicAdd(&gh[i], v);
  }
}

__global__ void scanKernel(unsigned* ws, int pass) {
  int s = threadIdx.x;
  if (s < 3) {
    const unsigned* hp = ws + WS_HIST + (pass * 3 + s) * 256;
    int shift = 8 * (3 - pass);
    unsigned need = ws[WS_KREM + s];
    unsigned cum = 0u;
    for (int b = 255; b >= 0; --b) {
      unsigned c = hp[b];
      if (cum + c >= need) {
        ws[WS_PREFIX + s] |= ((unsigned)b) << shift;
        ws[WS_KREM + s] = need - cum;
        break;
      }
      cum += c;
    }
  }
}

__global__ void compactKernel(const float* __restrict__ A, const int* __restrict__ blp,
                              unsigned* ws, int N) {
  int bl = blp[0] & 1;
  unsigned T[3] = {ws[WS_PREFIX + 0], ws[WS_PREFIX + 1], ws[WS_PREFIX + 2]};
  for (int i = blockIdx.x * blockDim.x + threadIdx.x; i < N; i += gridDim.x * blockDim.x) {
    float ain  = A[2 * i + bl];
    float aout = A[2 * i + (1 - bl)];
    unsigned keys[3];
    keys[0] = monoKey(ain);
    keys[1] = monoKey(-ain);
    keys[2] = monoKey(aout);
#pragma unroll
    for (int s = 0; s < 3; ++s) {
      if (keys[s] >= T[s]) {
        unsigned slot = atomicAdd(&ws[WS_CNT + s], 1u);
        if (slot < CAND_CAP) {
          unsigned* c = ws + WS_CAND + (s * CAND_CAP + slot) * 2;
          c[0] = keys[s];
          c[1] = (unsigned)i;
        }
      }
    }
  }
}

// One block per selection: bitonic sort 4096 (key desc, idx asc) composites in LDS.
__global__ void sortKernel(unsigned* ws, int K) {
  __shared__ unsigned long long arr[SORT_N];
  int s = blockIdx.x;
  int t = threadIdx.x;
  unsigned n = ws[WS_CNT + s];
  if (n > CAND_CAP) n = CAND_CAP;
  const unsigned* cand = ws + WS_CAND + s * CAND_CAP * 2;
  for (int i = t; i < SORT_N; i += blockDim.x) {
    unsigned long long v = 0ull;
    if (i < (int)n) {
      unsigned key = cand[2 * i], id = cand[2 * i + 1];
      v = (((unsigned long long)key) << 32) | (unsigned long long)(0xFFFFFFFFu - id);
    }
    arr[i] = v;
  }
  for (unsigned kk = 2; kk <= SORT_N; kk <<= 1) {
    for (unsigned j = kk >> 1; j > 0; j >>= 1) {
      __syncthreads();
      for (unsigned i = t; i < SORT_N; i += blockDim.x) {
        unsigned ixj = i ^ j;
        if (ixj > i) {
          bool up = ((i & kk) == 0u);
          unsigned long long a = arr[i], c = arr[ixj];
          bool sw = up ? (a < c) : (a > c);  // overall descending
          if (sw) { arr[i] = c; arr[ixj] = a; }
        }
      }
    }
  }
  __syncthreads();
  unsigned* sel = ws + WS_SEL + s * K;
  for (int i = t; i < K; i += blockDim.x)
    sel[i] = 0xFFFFFFFFu - (unsigned)(arr[i] & 0xFFFFFFFFull);
}

// One wave (32 threads) per 16-row tile. f32 WMMA 16x16x4 over K=D.
// A operand staged in LDS (row stride 516 floats: conflict-free b64 reads),
// W transposed in LDS so each lane's B pair is one contiguous b64 read.
// Lanes with col>=2 duplicate column col&1 (outputs never read) -> no masking.
#define A_STRIDE 516
__global__ void __launch_bounds__(32) gemmKernel(
    const float* __restrict__ h, const float* __restrict__ Wg,
    const float* __restrict__ bg, const unsigned* __restrict__ ws,
    float* __restrict__ out, int K, int D) {
  __shared__ float As[16 * A_STRIDE];  // 33 024 B
  __shared__ float Wt[2 * 512];        // 4 096 B  (Wt[c][k] = W[k][c])
  __shared__ unsigned rowIdxSh[16];

  int lane = threadIdx.x;
  int col  = lane & 15;
  int M3   = 3 * K;
  int r0   = blockIdx.x * 16;

  for (int i = lane; i < D; i += 32) {
    Wt[i]       = Wg[i * 2 + 0];
    Wt[512 + i] = Wg[i * 2 + 1];
  }
  if (lane < 16) {
    int row = r0 + lane;
    if (row >= M3) row = M3 - 1;
    int s = row / K, j = row - s * K;
    rowIdxSh[lane] = ws[WS_SEL + s * K + j];
  }
  __syncthreads();

  // Stage the 16 gathered rows: coalesced b128 global loads -> LDS.
  for (int r = 0; r < 16; ++r) {
    const float* src = h + (size_t)rowIdxSh[r] * (size_t)D;
#pragma unroll
    for (int c = lane * 4; c < D; c += 128) {
      __builtin_prefetch(src + c + 128, 0, 1);
      float4 v = *(const float4*)(src + c);
      *(float4*)(&As[r * A_STRIDE + c]) = v;
    }
  }
  __syncthreads();

  int kh = (lane >> 4) << 1;   // K sub-offset: lanes 0-15 -> 0, lanes 16-31 -> 2
  int cc = col & 1;            // duplicated class column for lanes col>=2
  const float* Arow = &As[col * A_STRIDE + kh];
  const float* Wrow = &Wt[cc * 512 + kh];

  v8f acc = {};
  for (int k = 0; k < D; k += 4) {
    v2f a    = *(const v2f*)(Arow + k);  // ds_load_b64
    v2f bvec = *(const v2f*)(Wrow + k);  // ds_load_b64 (broadcast)
    acc = __builtin_amdgcn_wmma_f32_16x16x4_f32(false, a, false, bvec,
                                                (short)0, acc, false, false);
  }

  float bias = bg[cc];
  bool colOK = (col < 2);
#pragma unroll
  for (int r = 0; r < 8; ++r) {
    int orow = r0 + r + ((lane >> 4) << 3);  // lanes 16-31 hold M = r+8
    float lu = acc[r] + bias;
    float other = __shfl_xor(lu, 1, 32);     // class 0 <-> class 1
    float m  = fmaxf(lu, other);
    float e  = expf(lu - m);
    float eo = expf(other - m);
    float p  = e / (e + eo);
    if (colOK && orow < M3) {
      out[M3 + orow * 2 + col]     = lu;   // logits_unnorm at [3K, 9K)
      out[3 * M3 + orow * 2 + col] = p;    // softmax at [9K, 15K)
    }
  }
  if (lane < 16) {
    int lr = r0 + lane;
    if (lr < M3) out[lr] = (lr < K) ? 1.0f : 0.0f;  // labels at [0, 3K)
  }
}

extern "C" void kernel_launch(void* const* d_in, const int* in_sizes, int n_in,
                              void* d_out, int out_size, void* d_ws, size_t ws_size,
                              hipStream_t stream) {
  const float* h  = (const float*)d_in[0];
  const float* A  = (const float*)d_in[1];
  const float* W  = (const float*)d_in[2];
  const float* b  = (const float*)d_in[3];
  const int*   bl = (const int*)d_in[4];

  int N = in_sizes[1] / 2;   // A is (N,1,2)
  int D = in_sizes[2] / 2;   // W is (D,2)
  int K = (int)(0.02 * (double)N);
  if (K == 0) K = 8;

  unsigned* ws = (unsigned*)d_ws;

  initKernel<<<1, 256, 0, stream>>>(ws, K);
  for (int p = 0; p < 4; ++p) {
    histKernel<<<512, 256, 0, stream>>>(A, bl, ws, N, p);
    scanKernel<<<1, 32, 0, stream>>>(ws, p);
  }
  compactKernel<<<512, 256, 0, stream>>>(A, bl, ws, N);
  sortKernel<<<3, 1024, 0, stream>>>(ws, K);

  int M3 = 3 * K;
  int blocks = (M3 + 15) / 16;
  gemmKernel<<<blocks, 32, 0, stream>>>(h, W, b, ws, (float*)d_out, K, D);
}